// Sage_33337536151789
// MI455X (gfx1250) — compile-verified
//
#include <hip/hip_runtime.h>

#define NNODES 50000
#define NFEAT  128
#define HIDDEN 64
#define NCLASS 40

typedef __attribute__((ext_vector_type(2))) float v2f;
typedef __attribute__((ext_vector_type(8))) float v8f;

__global__ void zero_f32(float* __restrict__ p, int n) {
    int i = blockIdx.x * blockDim.x + threadIdx.x;
    if (i < n) p[i] = 0.0f;
}

__global__ void degree_kernel(const int* __restrict__ dst, float* __restrict__ cnt, int E) {
    int e = blockIdx.x * blockDim.x + threadIdx.x;
    if (e < E) atomicAdd(&cnt[dst[e]], 1.0f);
}

// out[M, NDIM] = X[M, K] @ W[NDIM, K]^T  -- one wave per 16-row tile.
// fp32 WMMA 16x16x4: A lanes 0-15 hold K=0,1 / lanes 16-31 hold K=2,3 (2 VGPRs);
// B mirrored over N; C/D 8 VGPRs with M = v + 8*(lane>=16).
template <int K, int NDIM, int NT>
__global__ void gemm_nt_wmma(const float* __restrict__ X, const float* __restrict__ W,
                             float* __restrict__ out, int Mtiles) {
    int wave = blockIdx.x * (blockDim.x >> 5) + (threadIdx.x >> 5);
    if (wave >= Mtiles) return;               // uniform per wave: EXEC stays all-ones
    int lane  = threadIdx.x & 31;
    int lm    = lane & 15;
    int khalf = (lane >> 4) << 1;             // 0 or 2

    const float* xrow = X + (size_t)(wave * 16 + lm) * K + khalf;

    v8f acc[NT] = {};
#pragma unroll 4
    for (int k = 0; k < K; k += 4) {
        v2f a = *(const v2f*)(xrow + k);
#pragma unroll
        for (int t = 0; t < NT; ++t) {
            int n = t * 16 + lm;
            v2f b = {};
            if ((NDIM % 16 == 0) || (n < NDIM))
                b = *(const v2f*)(W + (size_t)n * K + k + khalf);
            acc[t] = __builtin_amdgcn_wmma_f32_16x16x4_f32(
                false, a, false, b, (short)0, acc[t], false, false);
        }
    }

    int mbase = wave * 16 + ((lane >> 4) << 3);
#pragma unroll
    for (int t = 0; t < NT; ++t) {
        int n = t * 16 + lm;
        if ((NDIM % 16 == 0) || (n < NDIM)) {
#pragma unroll
            for (int v = 0; v < 8; ++v)
                out[(size_t)(mbase + v) * NDIM + n] = acc[t][v];
        }
    }
}

// agg[dst[e], :] += feat[src[e], :], one thread per (edge, 4-float chunk)
template <int F>
__global__ void scatter_add_kernel(const float* __restrict__ feat, const int* __restrict__ src,
                                   const int* __restrict__ dst, float* __restrict__ agg, int E) {
    constexpr int C = F / 4;
    int tid = blockIdx.x * blockDim.x + threadIdx.x;
    if (tid >= E * C) return;
    int e = tid / C;
    int c = tid - e * C;
    int s = src[e], d = dst[e];
    float4 v = *((const float4*)(feat + (size_t)s * F) + c);
    float* base = agg + (size_t)d * F + c * 4;
    atomicAdd(base + 0, v.x);
    atomicAdd(base + 1, v.y);
    atomicAdd(base + 2, v.z);
    atomicAdd(base + 3, v.w);
}

// in-place: buf = act(buf / max(cnt,1) + bias[f] + self)
__global__ void combine_kernel(float* __restrict__ buf, const float* __restrict__ cnt,
                               const float* __restrict__ bias, const float* __restrict__ self,
                               int F, int total, int do_sigmoid) {
    int i = blockIdx.x * blockDim.x + threadIdx.x;
    if (i >= total) return;
    int n = i / F;
    int f = i - n * F;
    float c = fmaxf(cnt[n], 1.0f);
    float v = buf[i] / c + bias[f] + self[i];
    if (do_sigmoid) v = 1.0f / (1.0f + __expf(-v));
    buf[i] = v;
}

static inline int cdiv(int a, int b) { return (a + b - 1) / b; }

extern "C" void kernel_launch(void* const* d_in, const int* in_sizes, int n_in,
                              void* d_out, int out_size, void* d_ws, size_t ws_size,
                              hipStream_t stream) {
    const float* x   = (const float*)d_in[0];
    const int*   ei  = (const int*)d_in[1];
    const float* Wl1 = (const float*)d_in[2];
    const float* bl1 = (const float*)d_in[3];
    const float* Wr1 = (const float*)d_in[4];
    const float* Wl2 = (const float*)d_in[5];
    const float* bl2 = (const float*)d_in[6];
    const float* Wr2 = (const float*)d_in[7];
    float* out = (float*)d_out;

    const int E = in_sizes[1] / 2;
    const int* src = ei;
    const int* dst = ei + E;

    // workspace layout (floats): cnt[N] | Y[N*64] | Z[N*64] | AG[N*64]
    float* cnt = (float*)d_ws;
    float* Y   = cnt + NNODES;
    float* Z   = Y + (size_t)NNODES * HIDDEN;
    float* AG  = Z + (size_t)NNODES * HIDDEN;

    const int B  = 256;
    const int MT = NNODES / 16;  // 3125, exact

    // ---- layer 1 ----
    zero_f32<<<cdiv(NNODES, B), B, 0, stream>>>(cnt, NNODES);
    zero_f32<<<cdiv(NNODES * HIDDEN, B), B, 0, stream>>>(AG, NNODES * HIDDEN);
    degree_kernel<<<cdiv(E, B), B, 0, stream>>>(dst, cnt, E);

    gemm_nt_wmma<NFEAT, HIDDEN, 4><<<cdiv(MT, B / 32), B, 0, stream>>>(x, Wl1, Y, MT);
    gemm_nt_wmma<NFEAT, HIDDEN, 4><<<cdiv(MT, B / 32), B, 0, stream>>>(x, Wr1, Z, MT);

    scatter_add_kernel<HIDDEN><<<cdiv(E * (HIDDEN / 4), B), B, 0, stream>>>(Y, src, dst, AG, E);
    combine_kernel<<<cdiv(NNODES * HIDDEN, B), B, 0, stream>>>(
        AG, cnt, bl1, Z, HIDDEN, NNODES * HIDDEN, 1);   // AG now holds H

    // ---- layer 2 ----  (H = AG)
    gemm_nt_wmma<HIDDEN, NCLASS, 3><<<cdiv(MT, B / 32), B, 0, stream>>>(AG, Wl2, Y, MT);
    gemm_nt_wmma<HIDDEN, NCLASS, 3><<<cdiv(MT, B / 32), B, 0, stream>>>(AG, Wr2, Z, MT);

    zero_f32<<<cdiv(NNODES * NCLASS, B), B, 0, stream>>>(out, NNODES * NCLASS);
    scatter_add_kernel<NCLASS><<<cdiv(E * (NCLASS / 4), B), B, 0, stream>>>(Y, src, dst, out, E);
    combine_kernel<<<cdiv(NNODES * NCLASS, B), B, 0, stream>>>(
        out, cnt, bl2, Z, NCLASS, NNODES * NCLASS, 0);
}